// GCNLayer_70334384439537
// MI455X (gfx1250) — compile-verified
//
#include <hip/hip_runtime.h>

#define BB 8
#define NN 4096
#define EE 65536
#define DD 256

typedef __attribute__((ext_vector_type(2))) float v2f;
typedef __attribute__((ext_vector_type(8))) float v8f;

// ---------------------------------------------------------------------------
// Zero the aggregation buffer (must be re-done every call: graph replays).
// ---------------------------------------------------------------------------
__global__ __launch_bounds__(256) void zero_agg_kernel(float4* __restrict__ p, long n4) {
    long i = (long)blockIdx.x * blockDim.x + threadIdx.x;
    const long stride = (long)gridDim.x * blockDim.x;
    const float4 z = make_float4(0.f, 0.f, 0.f, 0.f);
    for (; i < n4; i += stride) p[i] = z;
}

// ---------------------------------------------------------------------------
// h = X @ W^T + b  via V_WMMA_F32_16X16X4_F32 (fp32 matrix pipe, exact dtype).
// Block = 256 threads (8 waves). Block tile: 16 rows x 256 cols.
// Wave w owns cols [w*32, w*32+32): two 16x16 f32 accumulators.
// A (16x4) layout: lane L -> row L%16, VGPR0/1 = K + 2*(L/16) + {0,1}.
// B (4x16)  layout: lane L -> col L%16, VGPR0/1 = K + 2*(L/16) + {0,1},
//   i.e. B[k][n] = W[n][k] -> contiguous float2 from row n of row-major W.
// C/D layout: VGPR r -> row r + 8*(L/16), col L%16.
// ---------------------------------------------------------------------------
__global__ __launch_bounds__(256) void gemm_wmma_kernel(
    const float* __restrict__ X, const float* __restrict__ W,
    const float* __restrict__ bias, float* __restrict__ H) {
    __shared__ float As[16][260];   // pad 256->260: bank = (m*4 + k) % 64, conflict-free

    const int row0 = blockIdx.x * 16;          // global output row base
    const int t    = threadIdx.x;

    // Stage the 16x256 X tile: each thread copies 16 contiguous floats.
    {
        const int r  = t >> 4;
        const int c0 = (t & 15) << 4;
        const float4* src = (const float4*)(X + (size_t)(row0 + r) * DD + c0);
        float4* dst = (float4*)(&As[r][c0]);   // (r*260 + c0) is 4-dword aligned
        dst[0] = src[0]; dst[1] = src[1]; dst[2] = src[2]; dst[3] = src[3];
    }
    __syncthreads();

    const int wave   = t >> 5;
    const int lane   = t & 31;
    const int laneHi = lane >> 4;   // 0 or 1
    const int laneLo = lane & 15;
    const int col0   = wave * 32;

    v8f acc0 = {};
    v8f acc1 = {};

    const float* wrow0 = W + (size_t)(col0 + laneLo) * DD;        // cols col0..col0+15
    const float* wrow1 = W + (size_t)(col0 + 16 + laneLo) * DD;   // cols col0+16..+31

    #pragma unroll 8
    for (int k0 = 0; k0 < DD; k0 += 4) {
        const int kk = k0 + laneHi * 2;
        const float2 av  = *(const float2*)(&As[laneLo][kk]);
        const float2 b0v = *(const float2*)(wrow0 + kk);
        const float2 b1v = *(const float2*)(wrow1 + kk);
        v2f a;  a.x  = av.x;  a.y  = av.y;
        v2f b0; b0.x = b0v.x; b0.y = b0v.y;
        v2f b1; b1.x = b1v.x; b1.y = b1v.y;
        acc0 = __builtin_amdgcn_wmma_f32_16x16x4_f32(false, a, false, b0,
                                                     (short)0, acc0, false, false);
        acc1 = __builtin_amdgcn_wmma_f32_16x16x4_f32(false, a, false, b1,
                                                     (short)0, acc1, false, false);
    }

    const int   c   = col0 + laneLo;
    const float bc0 = bias[c];
    const float bc1 = bias[c + 16];
    #pragma unroll
    for (int r = 0; r < 8; ++r) {
        const int m = r + laneHi * 8;
        float* o = H + (size_t)(row0 + m) * DD + c;
        o[0]  = acc0[r] + bc0;
        o[16] = acc1[r] + bc1;
    }
}

// ---------------------------------------------------------------------------
// Edge scatter: one wave per edge. agg[b,tgt,:] += h[b,src,:] * emask.
// Mask is wave-uniform -> whole-wave early exit, no divergence inside atomics.
// ---------------------------------------------------------------------------
__global__ __launch_bounds__(256) void scatter_kernel(
    const float* __restrict__ H, const long long* __restrict__ eidx,
    const int* __restrict__ mask, float* __restrict__ agg) {
    const int  wave = threadIdx.x >> 5;
    const int  lane = threadIdx.x & 31;
    const long slot = (long)blockIdx.x * 8 + wave;   // [0, B*E)
    const int  b    = (int)(slot >> 16);             // E = 65536
    const int  e    = (int)(slot & (EE - 1));

    const long long* ei = eidx + (size_t)b * 2 * EE; // (B,2,E) int64
    const int src = (int)ei[e];
    const int tgt = (int)ei[EE + e];

    const int* mb = mask + b * NN;
    if ((mb[src] & mb[tgt]) == 0) return;            // emask == 0 -> no contribution

    const float4* hs = (const float4*)(H + ((size_t)b * NN + src) * DD + lane * 8);
    float*        at = agg + ((size_t)b * NN + tgt) * DD + lane * 8;
    const float4 v0 = hs[0];
    const float4 v1 = hs[1];
    atomicAdd(at + 0, v0.x); atomicAdd(at + 1, v0.y);
    atomicAdd(at + 2, v0.z); atomicAdd(at + 3, v0.w);
    atomicAdd(at + 4, v1.x); atomicAdd(at + 5, v1.y);
    atomicAdd(at + 6, v1.z); atomicAdd(at + 7, v1.w);
}

// ---------------------------------------------------------------------------
// Row finalize: h += agg/sqrt(N); LayerNorm over D=256; gamma/beta; ReLU; mask.
// One wave32 per row, 8 elements per lane, shfl_xor tree reduction.
// ---------------------------------------------------------------------------
__global__ __launch_bounds__(256) void finalize_kernel(
    const float* __restrict__ H, const float* __restrict__ agg,
    const float* __restrict__ gamma, const float* __restrict__ beta,
    const int* __restrict__ mask, float* __restrict__ out) {
    const int  wave = threadIdx.x >> 5;
    const int  lane = threadIdx.x & 31;
    const long row  = (long)blockIdx.x * 8 + wave;   // [0, B*N)
    const int  base = lane * 8;

    const float4* hr = (const float4*)(H   + (size_t)row * DD + base);
    const float4* ar = (const float4*)(agg + (size_t)row * DD + base);
    const float4 h0 = hr[0], h1 = hr[1];
    const float4 a0 = ar[0], a1 = ar[1];

    const float inv = 0.015625f;                     // 1/sqrt(4096)
    float v[8];
    v[0] = h0.x + a0.x * inv; v[1] = h0.y + a0.y * inv;
    v[2] = h0.z + a0.z * inv; v[3] = h0.w + a0.w * inv;
    v[4] = h1.x + a1.x * inv; v[5] = h1.y + a1.y * inv;
    v[6] = h1.z + a1.z * inv; v[7] = h1.w + a1.w * inv;

    float s = 0.f, ss = 0.f;
    #pragma unroll
    for (int i = 0; i < 8; ++i) { s += v[i]; ss += v[i] * v[i]; }
    #pragma unroll
    for (int off = 16; off > 0; off >>= 1) {
        s  += __shfl_xor(s,  off, 32);
        ss += __shfl_xor(ss, off, 32);
    }
    const float mu   = s  * (1.0f / DD);
    const float var  = ss * (1.0f / DD) - mu * mu;
    const float rstd = rsqrtf(var + 1e-5f);
    const float fm   = (float)mask[row];

    const float4 g0 = *(const float4*)(gamma + base);
    const float4 g1 = *(const float4*)(gamma + base + 4);
    const float4 e0 = *(const float4*)(beta + base);
    const float4 e1 = *(const float4*)(beta + base + 4);
    const float g[8] = {g0.x, g0.y, g0.z, g0.w, g1.x, g1.y, g1.z, g1.w};
    const float be[8] = {e0.x, e0.y, e0.z, e0.w, e1.x, e1.y, e1.z, e1.w};

    float y[8];
    #pragma unroll
    for (int i = 0; i < 8; ++i) {
        const float t = (v[i] - mu) * rstd * g[i] + be[i];
        y[i] = fmaxf(t, 0.f) * fm;
    }
    float4* o = (float4*)(out + (size_t)row * DD + base);
    o[0] = make_float4(y[0], y[1], y[2], y[3]);
    o[1] = make_float4(y[4], y[5], y[6], y[7]);
}

// ---------------------------------------------------------------------------
extern "C" void kernel_launch(void* const* d_in, const int* in_sizes, int n_in,
                              void* d_out, int out_size, void* d_ws, size_t ws_size,
                              hipStream_t stream) {
    const float*     X     = (const float*)d_in[0];      // (B,N,D) f32
    const float*     W     = (const float*)d_in[1];      // (D,D)   f32
    const float*     bias  = (const float*)d_in[2];      // (D,)    f32
    const float*     gamma = (const float*)d_in[3];      // (D,)    f32
    const float*     beta  = (const float*)d_in[4];      // (D,)    f32
    const long long* eidx  = (const long long*)d_in[5];  // (B,2,E) i64
    const int*       mask  = (const int*)d_in[6];        // (B,N)   i32
    float*           out   = (float*)d_out;              // (B,N,D) f32

    float* h   = (float*)d_ws;                           // 32 MB
    float* agg = h + (size_t)BB * NN * DD;               // 32 MB

    const long n4 = (long)BB * NN * DD / 4;
    zero_agg_kernel<<<2048, 256, 0, stream>>>((float4*)agg, n4);
    gemm_wmma_kernel<<<BB * NN / 16, 256, 0, stream>>>(X, W, bias, h);
    scatter_kernel<<<BB * EE / 8, 256, 0, stream>>>(h, eidx, mask, agg);
    finalize_kernel<<<BB * NN / 8, 256, 0, stream>>>(h, agg, gamma, beta, mask, out);
}